// ConGraph_63513976373536
// MI455X (gfx1250) — compile-verified
//
#include <hip/hip_runtime.h>
#include <hip/hip_bf16.h>

typedef __attribute__((ext_vector_type(2))) float v2f;
typedef __attribute__((ext_vector_type(4))) float vf4;
typedef __attribute__((ext_vector_type(8))) float v8f;

#define NPTS 16384
#define DIM 64
#define NWAVES 8
#define MB 32            // rows per block (two 16-row WMMA subtiles)
#define FLT_BIG 3.402823466e+38f

// ---------------- Kernel 0: squared norms ----------------
__global__ __launch_bounds__(256) void sqnorm_kernel(const float* __restrict__ X,
                                                     float* __restrict__ sq) {
    int i = blockIdx.x * blockDim.x + threadIdx.x;
    if (i < NPTS) {
        const vf4* r = (const vf4*)(X + (size_t)i * DIM);
        float s = 0.f;
#pragma unroll
        for (int k = 0; k < DIM / 4; ++k) {
            vf4 v = r[k];
            s += v.x * v.x + v.y * v.y + v.z * v.z + v.w * v.w;
        }
        sq[i] = s;
    }
}

// ---------------- Kernel 1: WMMA distance + fused top-2 ----------------
__global__ __launch_bounds__(256) void knn_top2_kernel(const float* __restrict__ X,
                                                       const float* __restrict__ sq,
                                                       int2* __restrict__ idxOut) {
    const int i0     = blockIdx.x * MB;
    const int tid    = threadIdx.x;
    const int wave   = tid >> 5;
    const int lane   = tid & 31;
    const int laneN  = lane & 15;      // column / row-within-16 index
    const int laneHi = lane >> 4;      // 0 or 1
    const int koff   = laneHi * 2;     // K offset within k=4 chunk

    // Preload A fragments for both 16-row subtiles, pre-scaled by -2
    // A layout (16x4 f32): vgpr v, lane l  ->  M = l&15, K = v + 2*(l>=16)
    v2f a[2][16];
#pragma unroll
    for (int s = 0; s < 2; ++s) {
        const float* arow = X + (size_t)(i0 + s * 16 + laneN) * DIM;
#pragma unroll
        for (int kc = 0; kc < 16; ++kc) {
            a[s][kc].x = -2.0f * arow[kc * 4 + koff];
            a[s][kc].y = -2.0f * arow[kc * 4 + koff + 1];
        }
    }

    // Running per-lane top-2 (value, index) per (subtile, C-vgpr-row r)
    float bv1[2][8], bv2[2][8];
    int   bi1[2][8], bi2[2][8];
#pragma unroll
    for (int s = 0; s < 2; ++s)
#pragma unroll
        for (int r = 0; r < 8; ++r) {
            bv1[s][r] = FLT_BIG; bv2[s][r] = FLT_BIG;
            bi1[s][r] = 0x7fffffff; bi2[s][r] = 0x7fffffff;
        }

#pragma unroll 1
    for (int jt = wave; jt < NPTS / 16; jt += NWAVES) {
        const int j0 = jt * 16;
        // B layout (4x16 f32): vgpr v, lane l -> N = l&15, K = v + 2*(l>=16)
        // B[k][n] = X[j0+n][k]
        v2f b[16];
        const float* brow = X + (size_t)(j0 + laneN) * DIM;
#pragma unroll
        for (int kc = 0; kc < 16; ++kc) {
            b[kc].x = brow[kc * 4 + koff];
            b[kc].y = brow[kc * 4 + koff + 1];
        }
        const float sqn  = sq[j0 + laneN];
        const int   jcol = j0 + laneN;

#pragma unroll
        for (int s = 0; s < 2; ++s) {
            v8f c = {};
#pragma unroll
            for (int kc = 0; kc < 16; ++kc) {
                c = __builtin_amdgcn_wmma_f32_16x16x4_f32(
                        /*neg_a=*/false, a[s][kc], /*neg_b=*/false, b[kc],
                        /*c_mod=*/(short)0, c, /*reuse_a=*/false, /*reuse_b=*/false);
            }
            // c[r] = -2*dot(x_{i0+16s+r+8*laneHi}, x_{jcol}); val = sq[j] - 2*dot
#pragma unroll
            for (int r = 0; r < 8; ++r) {
                float val = sqn + c[r];
                bool lt1 = val < bv1[s][r];
                bool lt2 = val < bv2[s][r];
                bv2[s][r] = lt1 ? bv1[s][r] : (lt2 ? val : bv2[s][r]);
                bi2[s][r] = lt1 ? bi1[s][r] : (lt2 ? jcol : bi2[s][r]);
                bv1[s][r] = lt1 ? val  : bv1[s][r];
                bi1[s][r] = lt1 ? jcol : bi1[s][r];
            }
        }
    }

    // Cross-lane merge within each 16-lane half (masks 1,2,4,8 never cross halves)
#pragma unroll
    for (int m = 1; m <= 8; m <<= 1) {
#pragma unroll
        for (int s = 0; s < 2; ++s)
#pragma unroll
            for (int r = 0; r < 8; ++r) {
                float w1 = __shfl_xor(bv1[s][r], m);
                int   j1 = __shfl_xor(bi1[s][r], m);
                float w2 = __shfl_xor(bv2[s][r], m);
                int   j2 = __shfl_xor(bi2[s][r], m);
                bool o1 = (w1 < bv1[s][r]) || (w1 == bv1[s][r] && j1 < bi1[s][r]);
                float c1 = o1 ? bv1[s][r] : w1;  int ci1 = o1 ? bi1[s][r] : j1;
                float c2 = o1 ? w2 : bv2[s][r];  int ci2 = o1 ? j2 : bi2[s][r];
                bool o2 = (c1 < c2) || (c1 == c2 && ci1 < ci2);
                bv1[s][r] = o1 ? w1 : bv1[s][r];
                bi1[s][r] = o1 ? j1 : bi1[s][r];
                bv2[s][r] = o2 ? c1 : c2;
                bi2[s][r] = o2 ? ci1 : ci2;
            }
    }

    // Cross-wave merge through LDS: per row, per wave, a sorted top-2
    __shared__ float gV1[MB][NWAVES], gV2[MB][NWAVES];
    __shared__ int   gI1[MB][NWAVES], gI2[MB][NWAVES];
    if (laneN == 0) {
#pragma unroll
        for (int s = 0; s < 2; ++s)
#pragma unroll
            for (int r = 0; r < 8; ++r) {
                int row = s * 16 + r + 8 * laneHi;
                gV1[row][wave] = bv1[s][r];  gI1[row][wave] = bi1[s][r];
                gV2[row][wave] = bv2[s][r];  gI2[row][wave] = bi2[s][r];
            }
    }
    __syncthreads();

    if (tid < MB) {
        const int row = tid;
        float t1 = FLT_BIG, t2 = FLT_BIG;
        int   x1 = 0x7fffffff, x2 = 0x7fffffff;
#pragma unroll
        for (int w = 0; w < NWAVES; ++w) {
#pragma unroll
            for (int p = 0; p < 2; ++p) {
                float v = p ? gV2[row][w] : gV1[row][w];
                int   j = p ? gI2[row][w] : gI1[row][w];
                bool b1 = (v < t1) || (v == t1 && j < x1);
                bool b2 = (v < t2) || (v == t2 && j < x2);
                if (b1) { t2 = t1; x2 = x1; t1 = v; x1 = j; }
                else if (b2) { t2 = v; x2 = j; }
            }
        }
        idxOut[i0 + row] = make_int2(x1, x2);
    }
}

// ---------------- Kernel 2: stream the 1 GiB adjacency (NT stores) ----------------
__global__ __launch_bounds__(256) void fill_adj_kernel(const int2* __restrict__ idx,
                                                       float* __restrict__ out) {
    // One float4 per thread; NPTS/4 = 4096 quads per row
    int gid = blockIdx.x * blockDim.x + threadIdx.x;
    int row = gid >> 12;            // / 4096
    int j   = (gid & 4095) << 2;    // column of first element
    int2 nn = idx[row];
    vf4 v;
    v.x = (j     == nn.x || j     == nn.y) ? 1.0f : 0.0f;
    v.y = (j + 1 == nn.x || j + 1 == nn.y) ? 1.0f : 0.0f;
    v.z = (j + 2 == nn.x || j + 2 == nn.y) ? 1.0f : 0.0f;
    v.w = (j + 3 == nn.x || j + 3 == nn.y) ? 1.0f : 0.0f;
    __builtin_nontemporal_store(v, (vf4*)out + gid);
}

extern "C" void kernel_launch(void* const* d_in, const int* in_sizes, int n_in,
                              void* d_out, int out_size, void* d_ws, size_t ws_size,
                              hipStream_t stream) {
    const float* X = (const float*)d_in[0];
    float* out = (float*)d_out;

    float* sq  = (float*)d_ws;                         // 16384 floats = 64 KiB
    int2*  idx = (int2*)((char*)d_ws + 64 * 1024);     // 16384 int2   = 128 KiB

    sqnorm_kernel<<<NPTS / 256, 256, 0, stream>>>(X, sq);
    knn_top2_kernel<<<NPTS / MB, 256, 0, stream>>>(X, sq, idx);
    fill_adj_kernel<<<(NPTS / 4) * (NPTS / 256), 256, 0, stream>>>(idx, out);
}